// ArmRGBReg_66065186947321
// MI455X (gfx1250) — compile-verified
//
#include <hip/hip_runtime.h>
#include <float.h>

#define V_TOT 107778
#define NUP   8000
#define NLOW  8000
#define BATCH 8
#define TOPK  50
#define THR   0.01f
#define FARSQ (9999.0f * 9999.0f)
#define WAVES 4                      // waves per block (wave32)
#define NTILE (NUP / 16)             // 500 column tiles

typedef __attribute__((ext_vector_type(2))) float v2f;
typedef __attribute__((ext_vector_type(8))) float v8f;

// ---------------------------------------------------------------------------
// Prep: gather upper/lower points once into packed float4 {x,y,z,|p|^2}
// so the WMMA loop streams contiguous B128s out of L2.
// ---------------------------------------------------------------------------
__global__ __launch_bounds__(256)
void prep_kernel(const float* __restrict__ mesh,
                 const int* __restrict__ upper_idx,
                 const int* __restrict__ lower_idx,
                 float4* __restrict__ up, float4* __restrict__ low) {
    int t = blockIdx.x * blockDim.x + threadIdx.x;
    if (t < NUP) {
        int gi = upper_idx[t];
        float x = mesh[3 * gi + 0], y = mesh[3 * gi + 1], z = mesh[3 * gi + 2];
        up[t] = make_float4(x, y, z, x * x + y * y + z * z);
    } else if (t < NUP + NLOW) {
        int gi = lower_idx[t - NUP];
        float x = mesh[3 * gi + 0], y = mesh[3 * gi + 1], z = mesh[3 * gi + 2];
        low[t - NUP] = make_float4(x, y, z, x * x + y * y + z * z);
    }
}

// ---------------------------------------------------------------------------
// Main: one block = 16 lower rows. 4 waves, each wave WMMA-scans every 4th
// 16-col tile of upper points, keeping a per-row top-50 list in LDS.
// dist^2 tile via V_WMMA_F32_16X16X4_F32:
//   A row m = [-2lx,-2ly,-2lz,1], B col n = [ux,uy,uz,|u|^2], C = |l_m|^2.
// ---------------------------------------------------------------------------
__global__ __launch_bounds__(32 * WAVES)
void knn_loss_kernel(const float4* __restrict__ up,
                     const float4* __restrict__ low,
                     const float* __restrict__ rgb,
                     const int* __restrict__ upper_idx,
                     const int* __restrict__ lower_idx,
                     float* __restrict__ out) {
    __shared__ float4 s_low[16];
    __shared__ float  s_d[WAVES][16][16];
    __shared__ float  s_val[WAVES][16][TOPK];
    __shared__ int    s_col[WAVES][16][TOPK];

    const int tid  = threadIdx.x;
    const int wave = tid >> 5;
    const int lane = tid & 31;
    const int half = lane >> 4;      // 0: K={0,1} lanes, 1: K={2,3} lanes
    const int n    = lane & 15;      // column within tile / row M for A-frag
    const int rowBase = blockIdx.x * 16;

    if (wave == 0 && lane < 16) s_low[lane] = low[rowBase + lane];
    if (lane < 16) {
        #pragma unroll 1
        for (int k = 0; k < TOPK; ++k) {
            s_val[wave][lane][k] = FLT_MAX;
            s_col[wave][lane][k] = 0;
        }
    }
    __syncthreads();

    // A fragment: lane holds A[m = n][k0], A[m][k0+1], k0 = 2*half (constant).
    float4 lm = s_low[n];
    v2f afrag;
    if (half == 0) { afrag.x = -2.0f * lm.x; afrag.y = -2.0f * lm.y; }
    else           { afrag.x = -2.0f * lm.z; afrag.y = 1.0f;         }

    // C init: VGPR v, lanes 0-15 -> row v, lanes 16-31 -> row v+8 : |l_row|^2.
    v8f cinit;
    #pragma unroll
    for (int v = 0; v < 8; ++v) cinit[v] = s_low[v + 8 * half].w;

    float max_val  = FLT_MAX;
    int   max_slot = 0;

    for (int it = 0; it < NTILE / WAVES; ++it) {
        const int tile = wave + WAVES * it;

        float4 um = up[tile * 16 + n];                 // B128, L2-resident
        if (it + 1 < NTILE / WAVES)
            __builtin_prefetch(&up[(tile + WAVES) * 16 + n], 0, 0);

        // B fragment: lane holds B[k0][n], B[k0+1][n].
        v2f bfrag;
        if (half == 0) { bfrag.x = um.x; bfrag.y = um.y; }
        else           { bfrag.x = um.z; bfrag.y = um.w; }

        v8f d = __builtin_amdgcn_wmma_f32_16x16x4_f32(
            /*neg_a=*/false, afrag, /*neg_b=*/false, bfrag,
            /*c_mod=*/(short)0, cinit, /*reuse_a=*/false, /*reuse_b=*/false);

        // Mask by |l_x - u_x| < THR, dump masked dist^2 tile to LDS.
        const float ux = um.x;
        #pragma unroll
        for (int v = 0; v < 8; ++v) {
            int m = v + 8 * half;
            float dx = fabsf(s_low[m].x - ux);
            s_d[wave][m][n] = (dx < THR) ? d[v] : FARSQ;
        }
        __syncthreads();

        // Streaming top-K insertion: lanes 0-15 each own one row.
        if (lane < 16) {
            const int r = lane;
            #pragma unroll 1
            for (int j = 0; j < 16; ++j) {
                float val = s_d[wave][r][j];
                if (val < max_val) {
                    s_val[wave][r][max_slot] = val;
                    s_col[wave][r][max_slot] = tile * 16 + j;
                    float mv = -FLT_MAX; int ms = 0;
                    #pragma unroll 1
                    for (int k = 0; k < TOPK; ++k) {
                        float sv = s_val[wave][r][k];
                        if (sv > mv) { mv = sv; ms = k; }
                    }
                    max_val = mv; max_slot = ms;
                }
            }
        }
        __syncthreads();
    }

    // Merge the 4 per-wave top-50 lists (wave 0's list is the base).
    if (tid < 16) {
        const int r = tid;
        float mv = -FLT_MAX; int ms = 0;
        #pragma unroll 1
        for (int k = 0; k < TOPK; ++k) {
            float sv = s_val[0][r][k];
            if (sv > mv) { mv = sv; ms = k; }
        }
        #pragma unroll 1
        for (int w = 1; w < WAVES; ++w) {
            #pragma unroll 1
            for (int k = 0; k < TOPK; ++k) {
                float val = s_val[w][r][k];
                if (val < mv) {
                    s_val[0][r][ms] = val;
                    s_col[0][r][ms] = s_col[w][r][k];
                    float m2 = -FLT_MAX; int s2 = 0;
                    #pragma unroll 1
                    for (int q = 0; q < TOPK; ++q) {
                        float sv = s_val[0][r][q];
                        if (sv > m2) { m2 = sv; s2 = q; }
                    }
                    mv = m2; ms = s2;
                }
            }
        }
    }
    __syncthreads();

    // Phase 3: mean rgb over the 50 neighbors (order-invariant), squared error.
    for (int t = tid; t < 16 * BATCH * 3; t += 32 * WAVES) {
        int r  = t / (BATCH * 3);
        int bc = t - r * (BATCH * 3);
        int b  = bc / 3, c = bc - 3 * b;
        float acc = 0.0f;
        #pragma unroll 1
        for (int k = 0; k < TOPK; ++k) {
            int col = s_col[0][r][k];
            int u   = upper_idx[col];
            acc += rgb[((size_t)b * V_TOT + (size_t)u) * 3 + c];
        }
        float mean = acc * (1.0f / TOPK);
        int   li   = lower_idx[rowBase + r];
        float x    = rgb[((size_t)b * V_TOT + (size_t)li) * 3 + c];
        float diff = x - mean;
        out[((size_t)b * NLOW + (size_t)(rowBase + r)) * 3 + c] = diff * diff;
    }
}

// ---------------------------------------------------------------------------
extern "C" void kernel_launch(void* const* d_in, const int* in_sizes, int n_in,
                              void* d_out, int out_size, void* d_ws, size_t ws_size,
                              hipStream_t stream) {
    const float* mesh      = (const float*)d_in[0];   // (V,3) f32
    const float* rgb       = (const float*)d_in[1];   // (B,V,3) f32
    const int*   upper_idx = (const int*)d_in[2];     // (NU,) int
    const int*   lower_idx = (const int*)d_in[3];     // (NL,) int

    float4* up  = (float4*)d_ws;                      // NUP float4
    float4* low = up + NUP;                           // NLOW float4 (256 KB total)

    prep_kernel<<<(NUP + NLOW + 255) / 256, 256, 0, stream>>>(
        mesh, upper_idx, lower_idx, up, low);

    knn_loss_kernel<<<NLOW / 16, 32 * WAVES, 0, stream>>>(
        up, low, rgb, upper_idx, lower_idx, (float*)d_out);
}